// SparseMemory_78262894067955
// MI455X (gfx1250) — compile-verified
//
#include <hip/hip_runtime.h>

typedef float v2f __attribute__((ext_vector_type(2)));
typedef float v8f __attribute__((ext_vector_type(8)));

#define B_     32
#define DIN    1024
#define M_     16384
#define W_     512
#define K_     16
#define TILE_M 128
#define TILE_W 32
#define LDT    36   /* padded LDS row stride in floats: 144B = 9*16B (b128-aligned), */
                    /* bank stride 36 mod 64 -> conflict-free across 16 rows        */

// ---------------------------------------------------------------------------
// CDNA5 async memory->LDS (ASYNCcnt path), inline asm per CDNA5 ISA.
// VDST VGPR = LDS byte offset, VADDR pair = 64-bit global address (GV mode).
// ---------------------------------------------------------------------------
__device__ __forceinline__ void async_ld_b128(unsigned lds_off, unsigned long long gaddr) {
  asm volatile("global_load_async_to_lds_b128 %0, %1, off"
               :: "v"(lds_off), "v"(gaddr) : "memory");
}
__device__ __forceinline__ void wait_async0() {
  asm volatile("s_wait_asynccnt 0x0" ::: "memory");
}

// ---------------------------------------------------------------------------
// Kernel 0: read_keys = xi @ W_read + b_read   (32x1024 @ 1024x512)
// One wave per 16x16 output tile: 2 row-tiles x 32 col-tiles = 64 waves.
// ---------------------------------------------------------------------------
__global__ void k_proj(const float* __restrict__ xi, const float* __restrict__ Wr,
                       const float* __restrict__ br, float* __restrict__ keys) {
  const int wid  = (blockIdx.x * blockDim.x + threadIdx.x) >> 5;  // 0..63
  const int lane = threadIdx.x & 31;
  const int rt   = wid & 1;        // batch-row tile
  const int ct   = wid >> 1;       // col tile 0..31
  const int nloc = lane & 15;
  const int ksel = (lane < 16) ? 0 : 2;
  const int arow = rt * 16 + nloc;
  const int bcol = ct * 16 + nloc;

  v8f acc = {};
  for (int k = 0; k < DIN; k += 4) {
    const int k0 = k + ksel;
    v2f a, b;
    a.x = xi[(size_t)arow * DIN + k0];
    a.y = xi[(size_t)arow * DIN + k0 + 1];
    b.x = Wr[(size_t)k0 * W_ + bcol];
    b.y = Wr[(size_t)(k0 + 1) * W_ + bcol];
    acc = __builtin_amdgcn_wmma_f32_16x16x4_f32(false, a, false, b,
                                                (short)0, acc, false, false);
  }
  const float bias = br[bcol];
#pragma unroll
  for (int v = 0; v < 8; ++v) {
    const int row = rt * 16 + v + ((lane < 16) ? 0 : 8);
    keys[(size_t)row * W_ + bcol] = acc[v] + bias;
  }
}

// ---------------------------------------------------------------------------
// Kernel A: d2[b,m] = ||sparse[b,m,:]||^2 - 2*cross + ||key_b||^2
// Block: 256 threads (8 waves), 128 m-rows of one batch.
// Double-buffered async(memory->LDS) 128x32 tiles; WMMA f32 16x16x4 computes
// the key.row cross terms (16 rows/wave), VALU accumulates squares on the
// same LDS operands (co-executes with the matrix pipe).
// ---------------------------------------------------------------------------
__global__ void k_d2(const float* __restrict__ sparse, const float* __restrict__ keys,
                     float* __restrict__ d2) {
  __shared__ float buf[2][TILE_M * LDT];   // 2 x 18 KB
  __shared__ float keyL[W_];
  __shared__ float red[256];

  const int tid = threadIdx.x;
  const int b   = blockIdx.y;
  const int m0  = blockIdx.x * TILE_M;

  const int cg = tid & 7;    // async load: float4 column group (8 per row)
  const int rg = tid >> 3;   // async load: row group (0..31)
  const float* srcBase = sparse + ((size_t)b * M_ + m0) * W_;

  // issue chunk 0 (w0 = 0) async loads immediately; DMA overlaps the prologue
#pragma unroll
  for (int p = 0; p < 4; ++p) {
    const int r = rg + p * 32;
    const unsigned loff =
        (unsigned)(size_t)(const void*)&buf[0][r * LDT + cg * 4];
    async_ld_b128(loff,
        (unsigned long long)(size_t)(srcBase + (size_t)r * W_ + cg * 4));
  }

  // stage the 512-float read key, compute ||key||^2 via block reduce
  keyL[tid]       = keys[(size_t)b * W_ + tid];
  keyL[tid + 256] = keys[(size_t)b * W_ + tid + 256];
  __syncthreads();
  red[tid] = keyL[tid] * keyL[tid] + keyL[tid + 256] * keyL[tid + 256];
  __syncthreads();
  for (int off = 128; off > 0; off >>= 1) {
    if (tid < off) red[tid] += red[tid + off];
    __syncthreads();
  }
  const float sqkey = red[0];

  const int lane = tid & 31;
  const int wave = tid >> 5;
  const int ksel = (lane < 16) ? 0 : 2;
  const int nloc = lane & 15;

  v8f   acc = {};
  float sq  = 0.0f;

  wait_async0();     // chunk 0 landed in LDS
  __syncthreads();

  for (int c = 0; c < W_ / TILE_W; ++c) {         // 16 chunks of 32 cols
    const int cur = c & 1;
    // prefetch next chunk into the other buffer (async, no VGPR staging)
    if (c + 1 < W_ / TILE_W) {
      const int w1 = (c + 1) * TILE_W;
#pragma unroll
      for (int p = 0; p < 4; ++p) {
        const int r = rg + p * 32;
        const unsigned loff =
            (unsigned)(size_t)(const void*)&buf[1 - cur][r * LDT + cg * 4];
        async_ld_b128(loff,
            (unsigned long long)(size_t)(srcBase + (size_t)r * W_ + w1 + cg * 4));
      }
    }
    // compute on current chunk: 8 WMMA K=4 steps
    const int w0 = c * TILE_W;
    const float* tb = &buf[cur][(wave * 16 + nloc) * LDT];
#pragma unroll
    for (int kc = 0; kc < TILE_W; kc += 4) {
      const v2f a  = *(const v2f*)&keyL[w0 + kc + ksel];
      const v2f bb = *(const v2f*)&tb[kc + ksel];
      sq += bb.x * bb.x + bb.y * bb.y;
      acc = __builtin_amdgcn_wmma_f32_16x16x4_f32(false, a, false, bb,
                                                  (short)0, acc, false, false);
    }
    wait_async0();   // next chunk complete (per-wave ASYNCcnt)
    __syncthreads(); // all waves done reading buf[cur] before it is refilled
  }

  // A rows were the key replicated -> every D row holds cross; acc[0] lane n
  // is cross[local row n%16]. Squares are split lane n / lane n+16.
  const float sqrow = sq + __shfl_xor(sq, 16, 32);
  const float cross = acc[0];
  if (lane < 16) {
    const int row = m0 + wave * 16 + lane;
    d2[(size_t)b * M_ + row] = sqrow - 2.0f * cross + sqkey;
  }
}

// ---------------------------------------------------------------------------
// Kernel B: k=16 smallest d2 per batch (ascending, ties -> lower index),
// weights = d / d_max, positions out (float to d_out, int to ws).
// d2 is mutated (selected -> +inf) — safe: kernel A rewrites it every call.
// ---------------------------------------------------------------------------
__global__ void k_topk(float* __restrict__ d2, float* __restrict__ outW,
                       float* __restrict__ outP, int* __restrict__ posI) {
  __shared__ float rv[256];
  __shared__ int   ri[256];
  __shared__ float selv[K_];
  __shared__ int   seli[K_];

  const int tid = threadIdx.x;
  const int b   = blockIdx.x;
  float* base = d2 + (size_t)b * M_;

  for (int it = 0; it < K_; ++it) {
    float mv = __builtin_inff();
    int   mi = 0x7fffffff;
    for (int j = tid; j < M_; j += 256) {
      const float v = base[j];
      if (v < mv) { mv = v; mi = j; }
    }
    rv[tid] = mv; ri[tid] = mi;
    __syncthreads();
    for (int off = 128; off > 0; off >>= 1) {
      if (tid < off) {
        const float ov = rv[tid + off];
        const int   oi = ri[tid + off];
        if (ov < rv[tid] || (ov == rv[tid] && oi < ri[tid])) {
          rv[tid] = ov; ri[tid] = oi;
        }
      }
      __syncthreads();
    }
    if (tid == 0) {
      selv[it] = rv[0];
      seli[it] = ri[0];
      base[ri[0]] = __builtin_inff();
    }
    __syncthreads();
  }

  if (tid < K_) {
    const float mx = selv[K_ - 1];  // largest of the k ascending distances
    outW[b * K_ + tid] = selv[tid] / mx;
    outP[b * K_ + tid] = (float)seli[tid];
    posI[b * K_ + tid] = seli[tid];
  }
}

// ---------------------------------------------------------------------------
// Kernel C: gather read_vectors[b,i,:] = sparse[b, pos[b,i], :]
// ---------------------------------------------------------------------------
__global__ void k_gather(const float* __restrict__ sparse, const int* __restrict__ posI,
                         float* __restrict__ outRV) {
  const int i = blockIdx.x;        // 0..511  (b*16 + k)
  const int b = i >> 4;
  const int pos = posI[i];
  const float4* src = (const float4*)(sparse + ((size_t)b * M_ + pos) * W_);
  float4*       dst = (float4*)(outRV + (size_t)i * W_);
  dst[threadIdx.x] = src[threadIdx.x];   // 128 threads x float4 = 512 floats
}

// ---------------------------------------------------------------------------
extern "C" void kernel_launch(void* const* d_in, const int* in_sizes, int n_in,
                              void* d_out, int out_size, void* d_ws, size_t ws_size,
                              hipStream_t stream) {
  (void)in_sizes; (void)n_in; (void)out_size; (void)ws_size;

  const float* xi     = (const float*)d_in[0];
  const float* sparse = (const float*)d_in[1];
  const float* Wr     = (const float*)d_in[2];
  const float* br     = (const float*)d_in[3];
  // d_in[4..9] (write_* projections) are dead code in the reference; k == 16.

  float* ws   = (float*)d_ws;
  float* keys = ws;                               // 32*512      = 16384 floats
  float* d2   = ws + (B_ * W_);                   // 32*16384    = 524288 floats
  int*   posI = (int*)(ws + B_ * W_ + B_ * M_);   // 32*16 ints

  float* out   = (float*)d_out;
  float* outRV = out;                             // [32,16,512]
  float* outW  = out + (size_t)B_ * K_ * W_;      // [32,16]
  float* outP  = outW + B_ * K_;                  // [32,16] (positions as float)

  k_proj  <<<16,                      128, 0, stream>>>(xi, Wr, br, keys);
  k_d2    <<<dim3(M_ / TILE_M, B_),   256, 0, stream>>>(sparse, keys, d2);
  k_topk  <<<B_,                      256, 0, stream>>>(d2, outW, outP, posI);
  k_gather<<<B_ * K_,                 128, 0, stream>>>(sparse, posI, outRV);
}